// TransformerEncoderLayer_73933567033495
// MI455X (gfx1250) — compile-verified
//
#include <hip/hip_runtime.h>
#include <hip/hip_bf16.h>
#include <math.h>

#define B_    4
#define L_    128
#define D_    256
#define F_    1024
#define H_    8
#define DH_   32
#define R_    64
#define S_    16
#define KRP_  1024
#define EPS_  1e-5f
#define CUTOFF_ 10.0f

typedef __attribute__((ext_vector_type(2))) float v2f;
typedef __attribute__((ext_vector_type(8))) float v8f;
typedef __attribute__((ext_vector_type(4))) unsigned int v4u;
typedef __attribute__((ext_vector_type(4))) int v4i;
typedef __attribute__((ext_vector_type(8))) int v8i;

// D = A(16x4) * B(4x16) + C, full f32 (native-precision WMMA path on gfx1250)
static __device__ __forceinline__ v8f wmma4(v2f a, v2f b, v8f c) {
  return __builtin_amdgcn_wmma_f32_16x16x4_f32(false, a, false, b, (short)0, c,
                                               false, false);
}

static __device__ __forceinline__ float wave_sum(float v) {
#pragma unroll
  for (int m = 16; m > 0; m >>= 1) v += __shfl_xor(v, m, 32);
  return v;
}
static __device__ __forceinline__ float wave_max(float v) {
#pragma unroll
  for (int m = 16; m > 0; m >>= 1) v = fmaxf(v, __shfl_xor(v, m, 32));
  return v;
}

// ---------------------------------------------------------------------------
// TDM: 2-D tile load (rows x 16 f32 cols) from global into LDS.
// D# bitfields per CDNA5 ISA §8.3 (group 0) / §8.4 (group 1).
// ---------------------------------------------------------------------------
static __device__ __forceinline__ void tdm_load_tile_f32_16w(
    const float* gptr, unsigned lds_off, unsigned rows, unsigned tensor_d0,
    unsigned tensor_d1, unsigned stride0) {
  unsigned long long ga = (unsigned long long)(uintptr_t)gptr;
  v4u g0;
  g0[0] = 1u;                                    // count=1, is_restore=0
  g0[1] = lds_off;                               // lds_addr (bytes)
  g0[2] = (unsigned)(ga & 0xFFFFFFFFull);        // global_addr[31:0]
  g0[3] = (unsigned)((ga >> 32) & 0x1FFFFFFu) |  // global_addr[56:32]
          (2u << 30);                            // type=2 ("image")
  v8i g1;
  g1[0] = (int)(2u << 16);                       // wg_mask=0, data_size=4B
  g1[1] = (int)((tensor_d0 & 0xFFFFu) << 16);    // tensor_dim0[15:0]
  g1[2] = (int)((tensor_d0 >> 16) | ((tensor_d1 & 0xFFFFu) << 16));
  g1[3] = (int)((tensor_d1 >> 16) | (16u << 16));       // tile_dim0 = 16
  g1[4] = (int)(rows & 0xFFFFu);                        // tile_dim1 = rows
  g1[5] = (int)stride0;                                 // tensor_dim0_stride lo
  g1[6] = 0;                                            // stride0 hi, dim1_stride lo
  g1[7] = 0;
  v4i z4 = {};
#if defined(__clang_major__) && (__clang_major__ >= 23)
  v8i z8 = {};
  __builtin_amdgcn_tensor_load_to_lds(g0, g1, z4, z4, z8, 0);
#else
  __builtin_amdgcn_tensor_load_to_lds(g0, g1, z4, z4, 0);
#endif
  __builtin_amdgcn_s_wait_tensorcnt(0);
}

// ---------------------------------------------------------------------------
// Kernel 1: WQ[row][h][k] = sum_d src[row, h*32+d] * rp_proj_w[k, h*32+d]
// (row = b*L+l, k = rp feature index). GEMM M=512, N=1024, K=32 per head.
// ---------------------------------------------------------------------------
__global__ void k_wq(const float* __restrict__ src, const float* __restrict__ Wp,
                     float* __restrict__ WQ) {
  const int lane = threadIdx.x & 31;
  const int nt = blockIdx.x;   // 0..63
  const int mt = blockIdx.y;   // 0..31
  const int h  = blockIdx.z;   // 0..7
  const int r15 = lane & 15;
  const int kh  = (lane >> 4) * 2;
  const int hi8 = (lane >> 4) * 8;
  const int col = nt * 16 + r15;
  const float* arow = src + (size_t)(mt * 16 + r15) * D_ + h * DH_;
  const float* bcol = Wp + (size_t)col * D_ + h * DH_;
  __builtin_prefetch(bcol, 0, 0);
  v8f acc = {};
#pragma unroll
  for (int kk = 0; kk < 8; ++kk) {
    int k0 = kk * 4 + kh;
    v2f a, b;
    a[0] = arow[k0]; a[1] = arow[k0 + 1];
    b[0] = bcol[k0]; b[1] = bcol[k0 + 1];
    acc = wmma4(a, b, acc);
  }
#pragma unroll
  for (int i = 0; i < 8; ++i) {
    int r = mt * 16 + i + hi8;
    WQ[((size_t)r * H_ + h) * KRP_ + col] = acc[i];
  }
}

// ---------------------------------------------------------------------------
// Kernel 2: base scores  SB[b,h,l,m] = scale * q[b,l,h,:] . q[b,m,h,:]
// ---------------------------------------------------------------------------
__global__ void k_base_scores(const float* __restrict__ src,
                              float* __restrict__ SB) {
  const int lane = threadIdx.x & 31;
  const int nt = blockIdx.x;   // m tile 0..7
  const int mt = blockIdx.y;   // l tile 0..7
  const int bh = blockIdx.z;   // 0..31
  const int b = bh >> 3, h = bh & 7;
  const int r15 = lane & 15;
  const int kh  = (lane >> 4) * 2;
  const int hi8 = (lane >> 4) * 8;
  const float* arow = src + (size_t)(b * L_ + mt * 16 + r15) * D_ + h * DH_;
  const float* bcol = src + (size_t)(b * L_ + nt * 16 + r15) * D_ + h * DH_;
  v8f acc = {};
#pragma unroll
  for (int kk = 0; kk < 8; ++kk) {
    int k0 = kk * 4 + kh;
    v2f a, bb;
    a[0] = arow[k0];  a[1] = arow[k0 + 1];
    bb[0] = bcol[k0]; bb[1] = bcol[k0 + 1];
    acc = wmma4(a, bb, acc);
  }
  const float scale = 0.17677669529663687f;  // 1/sqrt(32)
#pragma unroll
  for (int i = 0; i < 8; ++i) {
    int r = mt * 16 + i + hi8;
    int c = nt * 16 + r15;
    SB[(((size_t)(b * H_ + h)) * L_ + r) * L_ + c] = scale * acc[i];
  }
}

// ---------------------------------------------------------------------------
// Kernel 3: per-(b,l) fused rp-scores + softmax + AV + residual + LayerNorm1.
// Wave w handles head h=w. T GEMM: M=128(m), N=128(h,s), K=64(r) via WMMA,
// with the s-contraction folded into a 16-lane shuffle reduction.
// ---------------------------------------------------------------------------
__global__ __launch_bounds__(256) void k_attn(
    const float* __restrict__ src, const float* __restrict__ rel_diss,
    const float* __restrict__ rel_dirs, const float* __restrict__ rp_b,
    const float* __restrict__ WQ, const float* __restrict__ SB,
    const float* __restrict__ g1, const float* __restrict__ be1,
    float* __restrict__ X) {
  __shared__ float s_drel[L_];
  __shared__ float s_fc[L_];
  __shared__ float s_rbf[L_ * R_];   // 32 KB
  __shared__ float s_sh[L_ * S_];    // 8 KB
  __shared__ float s_score[H_ * L_]; // 4 KB
  __shared__ float s_red[8];
  __shared__ float s_red2[8];

  const int t = threadIdx.x;
  const int lane = t & 31;
  const int w = t >> 5;          // wave id == head id
  const int bl = blockIdx.x;     // 0..511
  const int b = bl >> 7;
  const int l = bl & 127;

  // ---- stage distances, cosine cutoff, spherical harmonics ----
  if (t < L_) {
    float d = rel_diss[(size_t)bl * L_ + t];
    s_drel[t] = d;
    float u = fminf(fmaxf(d * (1.0f / CUTOFF_), 0.0f), 1.0f);
    s_fc[t] = 0.5f * (__cosf(3.14159265358979323846f * u) + 1.0f);
    const float* dir = rel_dirs + ((size_t)bl * L_ + t) * 3;
    float x = dir[0], y = dir[1], z = dir[2];
    float x2 = x * x, y2 = y * y, z2 = z * z;
    float* sp = s_sh + t * S_;
    sp[0]  = 0.28209479177387814f;
    sp[1]  = 0.4886025119029199f * y;
    sp[2]  = 0.4886025119029199f * z;
    sp[3]  = 0.4886025119029199f * x;
    sp[4]  = 1.0925484305920792f * x * y;
    sp[5]  = 1.0925484305920792f * y * z;
    sp[6]  = 0.31539156525252005f * (3.0f * z2 - 1.0f);
    sp[7]  = 1.0925484305920792f * x * z;
    sp[8]  = 0.5462742152960396f * (x2 - y2);
    sp[9]  = 0.5900435899266435f * y * (3.0f * x2 - y2);
    sp[10] = 2.890611442640554f * x * y * z;
    sp[11] = 0.4570457994644658f * y * (5.0f * z2 - 1.0f);
    sp[12] = 0.3731763325901154f * z * (5.0f * z2 - 3.0f);
    sp[13] = 0.4570457994644658f * x * (5.0f * z2 - 1.0f);
    sp[14] = 1.445305721320277f * z * (x2 - y2);
    sp[15] = 0.5900435899266435f * x * (x2 - 3.0f * y2);
  }
  __syncthreads();

  // ---- rbf: 128 x 64, gaussian * cutoff ----
  const float spacing = CUTOFF_ / (float)(R_ - 1);
  const float inv_sp = 1.0f / spacing;
#pragma unroll
  for (int e = 0; e < 32; ++e) {
    int idx = e * 256 + t;
    int m = idx >> 6, r = idx & 63;
    float u = (s_drel[m] - (float)r * spacing) * inv_sp;
    s_rbf[idx] = __expf(-0.5f * u * u) * s_fc[m];
  }

  // qb[h] = q . rp_proj_b slice (constant over m)
  const int h = w;
  float qb = src[(size_t)bl * D_ + h * DH_ + lane] * rp_b[h * DH_ + lane];
  qb = wave_sum(qb);
  __syncthreads();

  // ---- T GEMM (WMMA) + s-contraction ----
  const float* WQh = WQ + ((size_t)bl * H_ + h) * KRP_;
  const float* SBh = SB + (((size_t)(b * H_ + h)) * L_ + l) * L_;
  __builtin_prefetch(WQh, 0, 0);
  __builtin_prefetch(SBh, 0, 0);
  const int r15 = lane & 15;
  const int kh  = (lane >> 4) * 2;
  const int hi8 = (lane >> 4) * 8;
  for (int mt = 0; mt < 8; ++mt) {
    v8f acc = {};
    const float* arow = s_rbf + (size_t)(mt * 16 + r15) * R_;
#pragma unroll
    for (int kk = 0; kk < 16; ++kk) {
      int k0 = kk * 4 + kh;  // radial index
      v2f a, bb;
      a[0]  = arow[k0];             a[1]  = arow[k0 + 1];
      bb[0] = WQh[k0 * S_ + r15];   bb[1] = WQh[(k0 + 1) * S_ + r15];
      acc = wmma4(a, bb, acc);
    }
#pragma unroll
    for (int i = 0; i < 8; ++i) {
      int m = mt * 16 + i + hi8;
      float v = s_sh[m * S_ + r15] * acc[i];
      v += __shfl_xor(v, 1, 32);
      v += __shfl_xor(v, 2, 32);
      v += __shfl_xor(v, 4, 32);
      v += __shfl_xor(v, 8, 32);
      if (r15 == 0) s_score[h * L_ + m] = v + qb + SBh[m];
    }
  }

  // ---- softmax over m (wave-local: this wave wrote this head's row) ----
  float vals[4], mx = -1e30f;
#pragma unroll
  for (int j = 0; j < 4; ++j) {
    vals[j] = s_score[h * L_ + lane + j * 32];
    mx = fmaxf(mx, vals[j]);
  }
  mx = wave_max(mx);
  float se = 0.f;
#pragma unroll
  for (int j = 0; j < 4; ++j) { vals[j] = __expf(vals[j] - mx); se += vals[j]; }
  se = wave_sum(se);
  float inv = 1.0f / se;
#pragma unroll
  for (int j = 0; j < 4; ++j) s_score[h * L_ + lane + j * 32] = vals[j] * inv;

  // ---- attn_out feature t = h*32 + lane ----
  const float* vb = src + (size_t)(b * L_) * D_ + t;
  float o = 0.f;
  for (int m = 0; m < L_; ++m) o += s_score[h * L_ + m] * vb[(size_t)m * D_];

  // ---- residual + LayerNorm1 ----
  float xv = src[(size_t)bl * D_ + t] + o;
  float ws = wave_sum(xv);
  if (lane == 0) s_red[w] = ws;
  __syncthreads();
  float mu = 0.f;
#pragma unroll
  for (int j = 0; j < 8; ++j) mu += s_red[j];
  mu *= (1.0f / D_);
  float e = xv - mu;
  float ws2 = wave_sum(e * e);
  if (lane == 0) s_red2[w] = ws2;
  __syncthreads();
  float var = 0.f;
#pragma unroll
  for (int j = 0; j < 8; ++j) var += s_red2[j];
  var *= (1.0f / D_);
  X[(size_t)bl * D_ + t] = e * rsqrtf(var + EPS_) * g1[t] + be1[t];
}

// ---------------------------------------------------------------------------
// Kernel 4: FF1 = leaky_relu(X @ lin1_w + lin1_b), M=512 N=1024 K=256.
// 8 waves share one 16-col x 256-row lin1_w tile staged by the Tensor Data
// Mover into LDS (8x cut in global B traffic); each wave owns one M-tile.
// ---------------------------------------------------------------------------
__global__ __launch_bounds__(256) void k_ff1(const float* __restrict__ X,
                                             const float* __restrict__ W1,
                                             const float* __restrict__ b1,
                                             float* __restrict__ FF1) {
  __shared__ float lds_w[D_ * 16];  // 256 rows x 16 cols, 16 KB
  const int t = threadIdx.x;
  const int lane = t & 31;
  const int w = t >> 5;        // 0..7
  const int nt = blockIdx.x;   // 0..63 (N tile)
  const int mt = blockIdx.y * 8 + w;  // 0..31 (M tile, one per wave)
  const int r15 = lane & 15, kh = (lane >> 4) * 2, hi8 = (lane >> 4) * 8;
  const int col = nt * 16 + r15;

  if (__builtin_amdgcn_readfirstlane(w) == 0) {
    // one DMA per workgroup: tile = W1[0:256, nt*16 : nt*16+16]
    tdm_load_tile_f32_16w(W1 + nt * 16, (unsigned)(uintptr_t)&lds_w[0],
                          /*rows=*/D_, /*tensor_d0=*/F_, /*tensor_d1=*/D_,
                          /*stride0=*/F_);
  }
  __syncthreads();

  const float* arow = X + (size_t)(mt * 16 + r15) * D_;
  v8f acc = {};
  for (int kk = 0; kk < 64; ++kk) {
    int k0 = kk * 4 + kh;
    v2f a, bb;
    a[0]  = arow[k0];              a[1]  = arow[k0 + 1];
    bb[0] = lds_w[k0 * 16 + r15];  bb[1] = lds_w[(k0 + 1) * 16 + r15];
    acc = wmma4(a, bb, acc);
  }
  float bias = b1[col];
#pragma unroll
  for (int i = 0; i < 8; ++i) {
    int r = mt * 16 + i + hi8;
    float v = acc[i] + bias;
    FF1[(size_t)r * F_ + col] = (v > 0.f) ? v : 0.01f * v;
  }
}

// ---------------------------------------------------------------------------
// Kernel 5: out = LN2(X + FF1 @ lin2_w + lin2_b). One WG per 16 rows so the
// full 256-wide row is resident in LDS for the LayerNorm.
// ---------------------------------------------------------------------------
__global__ __launch_bounds__(256) void k_ff2_ln(
    const float* __restrict__ FF1, const float* __restrict__ W2,
    const float* __restrict__ b2, const float* __restrict__ X,
    const float* __restrict__ g2, const float* __restrict__ be2,
    float* __restrict__ out) {
  __shared__ float ybuf[16 * D_];  // 16 KB
  __shared__ float red[16 * 16];
  const int t = threadIdx.x, lane = t & 31, w = t >> 5;
  const int mt = blockIdx.x;  // 0..31
  const int r15 = lane & 15, kh = (lane >> 4) * 2, hi8 = (lane >> 4) * 8;
  for (int tl = 0; tl < 2; ++tl) {
    int nt = w * 2 + tl;  // 0..15
    int col = nt * 16 + r15;
    const float* arow = FF1 + (size_t)(mt * 16 + r15) * F_;
    __builtin_prefetch(W2 + (size_t)kh * D_ + col, 0, 0);
    v8f acc = {};
    for (int kk = 0; kk < 256; ++kk) {
      int k0 = kk * 4 + kh;
      v2f a, bb;
      a[0]  = arow[k0];                  a[1]  = arow[k0 + 1];
      bb[0] = W2[(size_t)k0 * D_ + col]; bb[1] = W2[(size_t)(k0 + 1) * D_ + col];
      acc = wmma4(a, bb, acc);
    }
    float bias = b2[col];
#pragma unroll
    for (int i = 0; i < 8; ++i) {
      int ri = i + hi8;
      ybuf[ri * D_ + col] = acc[i] + bias + X[(size_t)(mt * 16 + ri) * D_ + col];
    }
  }
  __syncthreads();
  const int rl = t >> 4, seg = t & 15;
  float ps = 0.f;
#pragma unroll
  for (int j = 0; j < 16; ++j) ps += ybuf[rl * D_ + seg * 16 + j];
  red[rl * 16 + seg] = ps;
  __syncthreads();
  float mu = 0.f;
#pragma unroll
  for (int j = 0; j < 16; ++j) mu += red[rl * 16 + j];
  mu *= (1.0f / D_);
  __syncthreads();
  float ps2 = 0.f;
#pragma unroll
  for (int j = 0; j < 16; ++j) {
    float e = ybuf[rl * D_ + seg * 16 + j] - mu;
    ps2 += e * e;
  }
  red[rl * 16 + seg] = ps2;
  __syncthreads();
  float var = 0.f;
#pragma unroll
  for (int j = 0; j < 16; ++j) var += red[rl * 16 + j];
  var *= (1.0f / D_);
  float rs = rsqrtf(var + EPS_);
#pragma unroll
  for (int j = 0; j < 16; ++j) {
    int c = seg * 16 + j;
    float e = ybuf[rl * D_ + c] - mu;
    out[(size_t)(mt * 16 + rl) * D_ + c] = e * rs * g2[c] + be2[c];
  }
}

extern "C" void kernel_launch(void* const* d_in, const int* in_sizes, int n_in,
                              void* d_out, int out_size, void* d_ws,
                              size_t ws_size, hipStream_t stream) {
  (void)in_sizes; (void)n_in; (void)out_size; (void)ws_size;
  const float* src      = (const float*)d_in[0];
  const float* rel_diss = (const float*)d_in[1];
  const float* rel_dirs = (const float*)d_in[2];
  const float* rp_w     = (const float*)d_in[3];
  const float* rp_b     = (const float*)d_in[4];
  const float* lin1_w   = (const float*)d_in[5];
  const float* lin1_b   = (const float*)d_in[6];
  const float* lin2_w   = (const float*)d_in[7];
  const float* lin2_b   = (const float*)d_in[8];
  const float* g1       = (const float*)d_in[9];
  const float* be1      = (const float*)d_in[10];
  const float* g2       = (const float*)d_in[11];
  const float* be2      = (const float*)d_in[12];

  float* ws  = (float*)d_ws;
  float* WQ  = ws;                                   // 4,194,304 f32 (16 MB)
  float* SB  = WQ + (size_t)B_ * L_ * H_ * KRP_;     //   524,288 f32 (2 MB)
  float* X   = SB + (size_t)B_ * H_ * L_ * L_;       //   131,072 f32
  float* FF1 = X + (size_t)B_ * L_ * D_;             //   524,288 f32 (2 MB)

  k_wq<<<dim3(64, 32, 8), 32, 0, stream>>>(src, rp_w, WQ);
  k_base_scores<<<dim3(8, 8, 32), 32, 0, stream>>>(src, SB);
  k_attn<<<dim3(512), 256, 0, stream>>>(src, rel_diss, rel_dirs, rp_b, WQ, SB,
                                        g1, be1, X);
  k_ff1<<<dim3(64, 4), 256, 0, stream>>>(X, lin1_w, lin1_b, FF1);
  k_ff2_ln<<<dim3(32), 256, 0, stream>>>(FF1, lin2_w, lin2_b, X, g2, be2,
                                         (float*)d_out);
}